// CSPN_20658792694708
// MI455X (gfx1250) — compile-verified
//
#include <hip/hip_runtime.h>
#include <hip/hip_bf16.h>

typedef __attribute__((ext_vector_type(2))) float v2f;
typedef __attribute__((ext_vector_type(8))) float v8f;

#define HH 512
#define WW 512
#define NB 8
#define HWsz ((size_t)HH * WW)

// ---------------------------------------------------------------------------
// Stage 1: affinity conv (WMMA f32 16x16x4, K=9 padded to 12 -> 3 steps)
// + per-pixel normalization, producing planar kernel buffer Kc[b][9][H][W].
//
//   A (16x4 f32): lane m (0..15) holds K=(2*half+v) in VGPR v  -> W[chan=m][k]
//   B (4x16 f32): lane n (0..15) holds K=(2*half+v) in VGPR v  -> im2col[k][pix n]
//   C (16x16 f32): lane n VGPR c = aff[chan c][pix n]  (rows 8..15 zero)
//
// All loads are branch-free: addresses clamped in-bounds, values zeroed by
// v_cndmask when out of bounds / in the K=9..11 padding slots.
// ---------------------------------------------------------------------------
__global__ __launch_bounds__(256) void cspn_affinity_wmma(
    const float* __restrict__ x,    // [8,1,512,512]
    const float* __restrict__ Wa,   // [8,1,3,3]
    const float* __restrict__ ba,   // [8]
    float* __restrict__ Kc)         // [8,9,512,512]
{
  const int tiles_per_row = WW / 16;
  int wave = (blockIdx.x * blockDim.x + threadIdx.x) >> 5;
  int lane = threadIdx.x & 31;
  int n    = lane & 15;     // pixel-in-tile (B/C column) AND channel row (A)
  int half = lane >> 4;     // K half select

  int tw = wave % tiles_per_row;
  int th = (wave / tiles_per_row) % HH;
  int b  = wave / (tiles_per_row * HH);
  int w0 = tw * 16;

  const float* xb = x + (size_t)b * HWsz;

  // (dy+1) and (dx+1) for k=0..8, 2 bits each, packed little-endian.
  // k = ky*3+kx (SAME 3x3 cross-correlation), dy=ky-1, dx=kx-1.
  const unsigned DYP = 0x2A540u;   // 0,0,0,1,1,1,2,2,2
  const unsigned DXP = 0x24924u;   // 0,1,2,0,1,2,0,1,2

  v2f av[3], bv[3];
#pragma unroll
  for (int kk = 0; kk < 3; ++kk) {
#pragma unroll
    for (int v = 0; v < 2; ++v) {
      int  k      = kk * 4 + half * 2 + v;   // global K slot 0..11
      bool kvalid = (k < 9);
      int  kc     = kvalid ? k : 0;          // clamp for safe indexing

      int dy = (int)((DYP >> (2 * kc)) & 3u) - 1;
      int dx = (int)((DXP >> (2 * kc)) & 3u) - 1;
      int hh = th + dy, wc = w0 + n + dx;
      bool inb = kvalid & ((unsigned)hh < (unsigned)HH)
                        & ((unsigned)wc < (unsigned)WW);
      int hcl = min(max(hh, 0), HH - 1);
      int wcl = min(max(wc, 0), WW - 1);

      float bval = xb[(size_t)hcl * WW + wcl];       // always in-bounds
      bval = inb ? bval : 0.0f;                      // v_cndmask, no branch

      float aval = Wa[(n & 7) * 9 + kc];             // always in-bounds
      aval = (kvalid && n < 8) ? aval : 0.0f;

      if (v == 0) { av[kk].x = aval; bv[kk].x = bval; }
      else        { av[kk].y = aval; bv[kk].y = bval; }
    }
  }

  v8f acc = {};
#pragma unroll
  for (int kk = 0; kk < 3; ++kk) {
    acc = __builtin_amdgcn_wmma_f32_16x16x4_f32(
        /*neg_a=*/false, av[kk], /*neg_b=*/false, bv[kk],
        /*c_mod=*/(short)0, acc, /*reuse_a=*/false, /*reuse_b=*/false);
  }

  // Normalization: lane n (0..15) owns all 8 channels of pixel (th, w0+n).
  if (lane < 16) {
    float a[8];
    float abs_sum = 0.0f;
#pragma unroll
    for (int c = 0; c < 8; ++c) {
      a[c] = acc[c] + ba[c];
      abs_sum += fabsf(a[c]);
    }
    float inv = 1.0f / abs_sum;
    float s = 0.0f;
#pragma unroll
    for (int c = 0; c < 8; ++c) { a[c] *= inv; s += a[c]; }

    float* Kb = Kc + (size_t)b * 9 * HWsz;
    size_t pix = (size_t)th * WW + (w0 + n);
    Kb[pix] = 1.0f - s;                     // channel 0 pairs with offset (0,0)
#pragma unroll
    for (int c = 0; c < 8; ++c) Kb[(size_t)(c + 1) * HWsz + pix] = a[c];
  }
}

// ---------------------------------------------------------------------------
// Stage 2: one diffusion iteration. Per-pixel 3x3 stencil with per-pixel
// weights. x tile staged in LDS with 1-pixel zero halo (9x reuse); the 9
// coefficient planes are read coalesced from the L2-resident planar buffer.
// Coefficient base is biased by +4 planes so all 9 plane offsets fit the
// signed 24-bit vmem instruction offset (-4MB..+4MB).
// ---------------------------------------------------------------------------
#define TX 64
#define TY 4

__global__ __launch_bounds__(256) void cspn_diffuse(
    const float* __restrict__ xin,
    const float* __restrict__ Kc,
    float* __restrict__ xout)
{
  __shared__ float tile[(TY + 2) * (TX + 2)];

  const int bpr = WW / TX;            // blocks per row
  const int bpc = HH / TY;            // block rows per image
  int bx = blockIdx.x % bpr;
  int by = (blockIdx.x / bpr) % bpc;
  int b  = blockIdx.x / (bpr * bpc);

  int tx = threadIdx.x, ty = threadIdx.y;
  int w = bx * TX + tx, h = by * TY + ty;

  const float* xb = xin + (size_t)b * HWsz;

  // Cooperative load of (TY+2)x(TX+2) tile, zero outside the image.
  for (int i = ty * TX + tx; i < (TY + 2) * (TX + 2); i += TX * TY) {
    int ly = i / (TX + 2), lx = i % (TX + 2);
    int gh = by * TY + ly - 1, gw = bx * TX + lx - 1;
    float v = (gh >= 0 && gh < HH && gw >= 0 && gw < WW)
                  ? xb[(size_t)gh * WW + gw] : 0.0f;
    tile[i] = v;
  }
  __syncthreads();

  // OFFS order from the reference: (0,0),(0,1),(0,-1),(1,0),(1,1),(1,-1),
  //                                (-1,0),(-1,1),(-1,-1)
  const int OI[9] = {0, 0, 0, 1, 1, 1, -1, -1, -1};
  const int OJ[9] = {0, 1, -1, 0, 1, -1, 0, 1, -1};

  size_t pix = (size_t)h * WW + w;
  // Bias by +4 planes: plane j sits at offset (j-4)*HWsz elements.
  const float* Kb4 = Kc + (size_t)b * 9 * HWsz + 4 * HWsz + pix;

  float out = 0.0f;
#pragma unroll
  for (int j = 0; j < 9; ++j) {
    // x[h - oi, w - oj] lives at tile[(ty+1-oi)][(tx+1-oj)]
    float xv = tile[(ty + 1 - OI[j]) * (TX + 2) + (tx + 1 - OJ[j])];
    out = fmaf(Kb4[(ptrdiff_t)(j - 4) * (ptrdiff_t)HWsz], xv, out);
  }
  xout[(size_t)b * HWsz + pix] = out;
}

// ---------------------------------------------------------------------------
// Host launcher: stage 1 once, then 24 ping-ponged diffusion launches.
// Workspace layout: [72 MB kernel coeffs][8 MB ping buffer]  (80 MB total).
// Iteration parity: it even -> ws ping, it odd -> d_out; it=23 ends in d_out.
// ---------------------------------------------------------------------------
extern "C" void kernel_launch(void* const* d_in, const int* in_sizes, int n_in,
                              void* d_out, int out_size, void* d_ws, size_t ws_size,
                              hipStream_t stream) {
  const float* x  = (const float*)d_in[0];   // [8,1,512,512]
  const float* Wa = (const float*)d_in[1];   // [8,1,3,3]
  const float* ba = (const float*)d_in[2];   // [8]
  float* out  = (float*)d_out;

  float* Kc   = (float*)d_ws;                       // 8*9*512*512 f32 = 72 MB
  float* ping = Kc + (size_t)NB * 9 * HWsz;         // 8*512*512 f32  =  8 MB

  // Stage 1: 8 batches * 512 rows * 32 tiles = 131072 waves, 8 waves/block.
  int waves  = NB * HH * (WW / 16);
  int blocks = waves / 8;
  cspn_affinity_wmma<<<blocks, 256, 0, stream>>>(x, Wa, ba, Kc);

  // Stage 2: 24 diffusion iterations.
  dim3 blk(TX, TY, 1);
  int grid = NB * (WW / TX) * (HH / TY);
  const float* src = x;
  for (int it = 0; it < 24; ++it) {
    float* dst = (it & 1) ? out : ping;
    cspn_diffuse<<<grid, blk, 0, stream>>>(src, Kc, dst);
    src = dst;
  }
  // it = 23 (odd) wrote to d_out.
}